// STGCN_84507776516518
// MI455X (gfx1250) — compile-verified
//
#include <hip/hip_runtime.h>

typedef float v2f __attribute__((ext_vector_type(2)));
typedef float v8f __attribute__((ext_vector_type(8)));

#define NN 113
#define INV_STD 0.9999950000374997f

__device__ __forceinline__ v8f wmma4(v2f a, v2f b, v8f c) {
  // D = A(16x4 f32) * B(4x16 f32) + C(16x16 f32)
  return __builtin_amdgcn_wmma_f32_16x16x4_f32(false, a, false, b, (short)0, c,
                                               false, false);
}

__device__ __forceinline__ float sigm(float x) {
  return 1.0f / (1.0f + __expf(-x));
}

// dst[co*K + kc] = src[kc*C + co]   (conv weights come in [kc][co] order)
__global__ void transpose_w(const float* __restrict__ src,
                            float* __restrict__ dst, int K, int C) {
  int i = blockIdx.x * blockDim.x + threadIdx.x;
  if (i < K * C) {
    int co = i / K;
    int kc = i - co * K;
    dst[i] = src[kc * C + co];
  }
}

// ---------------------------------------------------------------------------
// Gated temporal conv (kernel_t=3) with Cout=32, fused with 32x32 Cheb linear.
// One wave -> 16 output rows x 32 channels.  Row r = ((b*Tout)+t)*113 + n.
// ---------------------------------------------------------------------------
template <int CIN>
__global__ __launch_bounds__(128) void gated_tcn_cheb(
    const float* __restrict__ X, int Tin, const float* __restrict__ wtP,
    const float* __restrict__ wtQ, const float* __restrict__ wtR,
    const float* __restrict__ bp, const float* __restrict__ bq,
    const float* __restrict__ br, const float* __restrict__ cwT,
    const float* __restrict__ cb, float* __restrict__ Out, int Mtiles) {
  constexpr int K = 3 * CIN;
  const int Tout = Tin - 2;
  __shared__ float tr[4][16][34];  // per-wave 16x32 transpose scratch (+pad)

  const int lane = threadIdx.x & 31;
  const int wv = threadIdx.x >> 5;
  const int lo = lane & 15;
  const int hi = lane >> 4;
  const int gw = blockIdx.x * 4 + wv;
  const int nw = gridDim.x * 4;

  for (int tile = gw; tile < Mtiles; tile += nw) {
    int r = tile * 16 + lo;  // this lane's A-matrix row
    int b = r / (Tout * NN);
    int rem = r - b * (Tout * NN);
    int t = rem / NN;
    int n = rem - t * NN;
    const float* arow = X + (((size_t)b * Tin + t) * NN + n) * CIN;

    v8f accP[2], accQ[2], accR[2];
#pragma unroll
    for (int j = 0; j < 2; ++j) {
      accP[j] = 0.0f;
      accQ[j] = 0.0f;
      accR[j] = 0.0f;
    }

    for (int kk = 0; kk < K / 4; ++kk) {
      int kc0 = kk * 4 + hi * 2;       // K-pair for this lane half
      int k = kc0 / CIN;               // temporal tap
      int ci = kc0 % CIN;              // input channel (even -> pair contiguous)
      v2f a = *(const v2f*)(arow + (size_t)k * (NN * CIN) + ci);
#pragma unroll
      for (int j = 0; j < 2; ++j) {
        int co = j * 16 + lo;
        v2f bP = *(const v2f*)(wtP + co * K + kc0);
        v2f bQ = *(const v2f*)(wtQ + co * K + kc0);
        v2f bR = *(const v2f*)(wtR + co * K + kc0);
        accP[j] = wmma4(a, bP, accP[j]);
        accQ[j] = wmma4(a, bQ, accQ[j]);
        accR[j] = wmma4(a, bR, accR[j]);
      }
    }

    // gating: relu(P * sigmoid(Q) + R), then write D-layout tile to LDS
    // D layout: element v -> row (v + 8*hi), col (j*16 + lo)
#pragma unroll
    for (int j = 0; j < 2; ++j) {
      int co = j * 16 + lo;
      float bP = bp[co], bQ = bq[co], bR = br[co];
#pragma unroll
      for (int v = 0; v < 8; ++v) {
        float p = accP[j][v] + bP;
        float q = accQ[j][v] + bQ;
        float rr = accR[j][v] + bR;
        tr[wv][v + 8 * hi][co] = fmaxf(fmaf(p, sigm(q), rr), 0.0f);
      }
    }
    __builtin_amdgcn_wave_barrier();

    // Cheb linear: (16x32) @ cw(32x32), read A-fragments from LDS transpose
    v8f acc2[2];
    acc2[0] = 0.0f;
    acc2[1] = 0.0f;
#pragma unroll
    for (int kk = 0; kk < 8; ++kk) {
      int kc0 = kk * 4 + hi * 2;
      v2f a = *(const v2f*)(&tr[wv][lo][kc0]);
#pragma unroll
      for (int j = 0; j < 2; ++j) {
        v2f bC = *(const v2f*)(cwT + (j * 16 + lo) * 32 + kc0);
        acc2[j] = wmma4(a, bC, acc2[j]);
      }
    }

#pragma unroll
    for (int j = 0; j < 2; ++j) {
      int co = j * 16 + lo;
      float bc = cb[co];
#pragma unroll
      for (int v = 0; v < 8; ++v) {
        int rr = tile * 16 + v + 8 * hi;
        Out[(size_t)rr * 32 + co] = fmaxf(acc2[j][v] + bc, 0.0f);
      }
    }
    __builtin_amdgcn_wave_barrier();
  }
}

// ---------------------------------------------------------------------------
// Gated temporal conv fused with eval-mode BatchNorm (per node) + outer ReLU.
// One wave -> 16 rows x 16 channels (j-tile in grid).
// ---------------------------------------------------------------------------
template <int CIN, int COUT>
__global__ __launch_bounds__(128) void gated_tcn_bn(
    const float* __restrict__ X, int Tin, const float* __restrict__ wtP,
    const float* __restrict__ wtQ, const float* __restrict__ wtR,
    const float* __restrict__ bp, const float* __restrict__ bq,
    const float* __restrict__ br, const float* __restrict__ gv,
    const float* __restrict__ bt, float* __restrict__ Out, int Mtiles) {
  constexpr int K = 3 * CIN;
  constexpr int JT = COUT / 16;
  const int Tout = Tin - 2;

  const int lane = threadIdx.x & 31;
  const int wv = threadIdx.x >> 5;
  const int lo = lane & 15;
  const int hi = lane >> 4;
  const int gw = blockIdx.x * 4 + wv;
  const int nw = gridDim.x * 4;
  const int total = Mtiles * JT;

  for (int tile = gw; tile < total; tile += nw) {
    int j = tile / Mtiles;
    int mt = tile - j * Mtiles;
    int r = mt * 16 + lo;
    int b = r / (Tout * NN);
    int rem = r - b * (Tout * NN);
    int t = rem / NN;
    int n = rem - t * NN;
    const float* arow = X + (((size_t)b * Tin + t) * NN + n) * CIN;
    const int co = j * 16 + lo;

    v8f accP, accQ, accR;
    accP = 0.0f;
    accQ = 0.0f;
    accR = 0.0f;

    for (int kk = 0; kk < K / 4; ++kk) {
      int kc0 = kk * 4 + hi * 2;
      int k = kc0 / CIN;
      int ci = kc0 % CIN;
      v2f a = *(const v2f*)(arow + (size_t)k * (NN * CIN) + ci);
      v2f bP = *(const v2f*)(wtP + co * K + kc0);
      v2f bQ = *(const v2f*)(wtQ + co * K + kc0);
      v2f bR = *(const v2f*)(wtR + co * K + kc0);
      accP = wmma4(a, bP, accP);
      accQ = wmma4(a, bQ, accQ);
      accR = wmma4(a, bR, accR);
    }

    float bP = bp[co], bQ = bq[co], bR = br[co];
#pragma unroll
    for (int v = 0; v < 8; ++v) {
      float p = accP[v] + bP;
      float q = accQ[v] + bQ;
      float rr = accR[v] + bR;
      float g = fmaxf(fmaf(p, sigm(q), rr), 0.0f);  // gated TCN output
      int orow = mt * 16 + v + 8 * hi;
      int orem = orow % (Tout * NN);
      int n2 = orem % NN;
      float o = fmaxf(fmaf(g, INV_STD * gv[n2], bt[n2]), 0.0f);  // BN + ReLU
      Out[(size_t)orow * COUT + co] = o;
    }
  }
}

// ---------------------------------------------------------------------------
// 113 per-node MLPs: feat(B,32) -> relu(Linear 32->64) -> Linear 64->3.
// feat[b, t*16+c] = h4[b,t,n,c].  One wave -> 16 batch rows of one node.
// ---------------------------------------------------------------------------
__global__ __launch_bounds__(128) void node_mlp(
    const float* __restrict__ H4, const float* __restrict__ w1,
    const float* __restrict__ b1, const float* __restrict__ w2,
    const float* __restrict__ b2, float* __restrict__ Out) {
  __shared__ float tr[4][16][66];  // 16x64 hidden transpose scratch (+pad)

  const int lane = threadIdx.x & 31;
  const int wv = threadIdx.x >> 5;
  const int lo = lane & 15;
  const int hi = lane >> 4;
  const int gw = blockIdx.x * 4 + wv;
  const int nw = gridDim.x * 4;
  const int TOT = NN * 64;  // 113 nodes * (1024/16) batch tiles

  for (int tile = gw; tile < TOT; tile += nw) {
    int n = tile / 64;
    int mt = tile - n * 64;
    int bidx = mt * 16 + lo;

    // Layer 1: (16x32) @ w1[n]^T (32x64)
    v8f acc1[4];
#pragma unroll
    for (int j = 0; j < 4; ++j) acc1[j] = 0.0f;

#pragma unroll
    for (int kk = 0; kk < 8; ++kk) {
      int kc0 = kk * 4 + hi * 2;
      int t = kc0 >> 4;
      int c = kc0 & 15;
      v2f a = *(const v2f*)(H4 + (((size_t)bidx * 2 + t) * NN + n) * 16 + c);
#pragma unroll
      for (int j = 0; j < 4; ++j) {
        int co = j * 16 + lo;
        v2f bW = *(const v2f*)(w1 + ((size_t)n * 64 + co) * 32 + kc0);
        acc1[j] = wmma4(a, bW, acc1[j]);
      }
    }

#pragma unroll
    for (int j = 0; j < 4; ++j) {
      int co = j * 16 + lo;
      float bb = b1[n * 64 + co];
#pragma unroll
      for (int v = 0; v < 8; ++v)
        tr[wv][v + 8 * hi][co] = fmaxf(acc1[j][v] + bb, 0.0f);
    }
    __builtin_amdgcn_wave_barrier();

    // Layer 2: (16x64) @ w2[n]^T (64x3), cols >= 3 padded with zero B
    v8f acc2;
    acc2 = 0.0f;
    int col = lo;
    int cc = col < 3 ? col : 2;
    v2f zero2;
    zero2 = 0.0f;
#pragma unroll
    for (int kk = 0; kk < 16; ++kk) {
      int kc0 = kk * 4 + hi * 2;
      v2f a = *(const v2f*)(&tr[wv][lo][kc0]);
      v2f bW = *(const v2f*)(w2 + ((size_t)n * 3 + cc) * 64 + kc0);
      bW = (col < 3) ? bW : zero2;
      acc2 = wmma4(a, bW, acc2);
    }

    if (col < 3) {
      float bb = b2[n * 3 + col];
#pragma unroll
      for (int v = 0; v < 8; ++v) {
        int bi = mt * 16 + v + 8 * hi;
        Out[((size_t)bi * NN + n) * 3 + col] = acc2[v] + bb;
      }
    }
    __builtin_amdgcn_wave_barrier();
  }
}

// ---------------------------------------------------------------------------
extern "C" void kernel_launch(void* const* d_in, const int* in_sizes, int n_in,
                              void* d_out, int out_size, void* d_ws,
                              size_t ws_size, hipStream_t stream) {
  (void)in_sizes; (void)n_in; (void)out_size; (void)ws_size;
  const float* x = (const float*)d_in[0];
  // d_in[1] = edge_index (unused: ChebConv K=1 identity term only)
  const float* a_wp = (const float*)d_in[2];
  const float* a_bp = (const float*)d_in[3];
  const float* a_wq = (const float*)d_in[4];
  const float* a_bq = (const float*)d_in[5];
  const float* a_wr = (const float*)d_in[6];
  const float* a_br = (const float*)d_in[7];
  const float* a_cw = (const float*)d_in[8];
  const float* a_cb = (const float*)d_in[9];
  const float* a_wp2 = (const float*)d_in[10];
  const float* a_bp2 = (const float*)d_in[11];
  const float* a_wq2 = (const float*)d_in[12];
  const float* a_bq2 = (const float*)d_in[13];
  const float* a_wr2 = (const float*)d_in[14];
  const float* a_br2 = (const float*)d_in[15];
  const float* a_g = (const float*)d_in[16];
  const float* a_beta = (const float*)d_in[17];
  const float* b_wp = (const float*)d_in[18];
  const float* b_bp = (const float*)d_in[19];
  const float* b_wq = (const float*)d_in[20];
  const float* b_bq = (const float*)d_in[21];
  const float* b_wr = (const float*)d_in[22];
  const float* b_br = (const float*)d_in[23];
  const float* b_cw = (const float*)d_in[24];
  const float* b_cb = (const float*)d_in[25];
  const float* b_wp2 = (const float*)d_in[26];
  const float* b_bp2 = (const float*)d_in[27];
  const float* b_wq2 = (const float*)d_in[28];
  const float* b_bq2 = (const float*)d_in[29];
  const float* b_wr2 = (const float*)d_in[30];
  const float* b_br2 = (const float*)d_in[31];
  const float* b_g = (const float*)d_in[32];
  const float* b_beta = (const float*)d_in[33];
  const float* mlp_w1 = (const float*)d_in[34];
  const float* mlp_b1 = (const float*)d_in[35];
  const float* mlp_w2 = (const float*)d_in[36];
  const float* mlp_b2 = (const float*)d_in[37];

  float* ws = (float*)d_ws;
  float* h1 = ws;                       // (1024,8,113,32)  29,622,272 f
  float* h2 = ws + 29622272;            // (1024,6,113,64)  44,433,408 f
  float* h3 = ws;                       // reuse: (1024,4,113,32)
  float* h4 = ws + 29622272;            // reuse: (1024,2,113,16)
  float* wt = ws + 29622272 + 44433408; // transposed weights (44,672 f)

  auto T = [&](const float* s, float* d, int K, int C) {
    transpose_w<<<(K * C + 255) / 256, 256, 0, stream>>>(s, d, K, C);
  };
  T(a_wp, wt + 0, 12, 32);
  T(a_wq, wt + 384, 12, 32);
  T(a_wr, wt + 768, 12, 32);
  T(a_cw, wt + 1152, 32, 32);
  T(a_wp2, wt + 2176, 96, 64);
  T(a_wq2, wt + 8320, 96, 64);
  T(a_wr2, wt + 14464, 96, 64);
  T(b_wp, wt + 20608, 192, 32);
  T(b_wq, wt + 26752, 192, 32);
  T(b_wr, wt + 32896, 192, 32);
  T(b_cw, wt + 39040, 32, 32);
  T(b_wp2, wt + 40064, 96, 16);
  T(b_wq2, wt + 41600, 96, 16);
  T(b_wr2, wt + 43136, 96, 16);

  // Block 1: gated TCN (4->32,K=12) + Cheb 32x32   (M = 1024*8*113)
  gated_tcn_cheb<4><<<2048, 128, 0, stream>>>(
      x, 10, wt + 0, wt + 384, wt + 768, a_bp, a_bq, a_br, wt + 1152, a_cb, h1,
      57856);
  // Block 1: gated TCN (32->64,K=96) + BN + ReLU   (M = 1024*6*113)
  gated_tcn_bn<32, 64><<<2048, 128, 0, stream>>>(
      h1, 8, wt + 2176, wt + 8320, wt + 14464, a_bp2, a_bq2, a_br2, a_g,
      a_beta, h2, 43392);
  // Block 2: gated TCN (64->32,K=192) + Cheb 32x32 (M = 1024*4*113)
  gated_tcn_cheb<64><<<2048, 128, 0, stream>>>(
      h2, 6, wt + 20608, wt + 26752, wt + 32896, b_bp, b_bq, b_br, wt + 39040,
      b_cb, h3, 28928);
  // Block 2: gated TCN (32->16,K=96) + BN + ReLU   (M = 1024*2*113)
  gated_tcn_bn<32, 16><<<2048, 128, 0, stream>>>(
      h3, 4, wt + 40064, wt + 41600, wt + 43136, b_bp2, b_bq2, b_br2, b_g,
      b_beta, h4, 14464);
  // Per-node MLPs
  node_mlp<<<1024, 128, 0, stream>>>(h4, mlp_w1, mlp_b1, mlp_w2, mlp_b2,
                                     (float*)d_out);
}